// SpikeFP64SinCos_43860206027306
// MI455X (gfx1250) — compile-verified
//
#include <hip/hip_runtime.h>
#include <cstdint>

#define THREADS     256
#define TILE_ELEMS  256          // elements per block
#define TILE_BYTES  65536        // 256 elems * 64 pulses * 4B

// ---------------- gfx1250 async global<->LDS DMA helpers ----------------
__device__ __forceinline__ void async_load16(uint32_t lds_off, uint64_t gaddr) {
  asm volatile("global_load_async_to_lds_b128 %0, %1, off"
               :: "v"(lds_off), "v"(gaddr) : "memory");
}
__device__ __forceinline__ void async_store16(uint64_t gaddr, uint32_t lds_off) {
  asm volatile("global_store_async_from_lds_b128 %0, %1, off"
               :: "v"(gaddr), "v"(lds_off) : "memory");
}
__device__ __forceinline__ void wait_async0() {
  asm volatile("s_wait_asynccnt 0" ::: "memory");
}

// ---------------- exact FP64 sincos core (bit-identical to reference) ----
__device__ __forceinline__ void sincos_fp64_bits(uint64_t xb, uint64_t& sb, uint64_t& cb) {
#pragma clang fp contract(off)
  constexpr double PI          = 3.141592653589793;   // == math.pi bits
  constexpr double TWO_OVER_PI = 2.0 / PI;            // IEEE-RN constant fold == Python
  constexpr double PIO2        = PI / 2.0;            // exact
  constexpr double S3 = -1.0 / 6.0,  S5 = 1.0 / 120.0, S7 = -1.0 / 5040.0;
  constexpr double C2 = -0.5,        C4 = 1.0 / 24.0,  C6 = -1.0 / 720.0;
  constexpr uint64_t SGN = 0x8000000000000000ull;

  const double   x   = __longlong_as_double((long long)xb);
  const double   xs  = x * TWO_OVER_PI;
  const uint64_t xsb = (uint64_t)__double_as_longlong(xs);
  const uint64_t qs  = xsb & SGN;                       // sign of q (kept from xs)
  const double   axs = __longlong_as_double((long long)(xsb & ~SGN));
  const double   qm  = __builtin_floor(axs + 0.5);      // |q|
  const double   q   = __longlong_as_double((long long)((uint64_t)__double_as_longlong(qm) | qs));
  const double   t   = q * PIO2;
  const double   mt  = __longlong_as_double((long long)((uint64_t)__double_as_longlong(t) ^ SGN));
  const double   r   = x + mt;                          // adder with sign-flipped q*pi/2

  const double r2 = r * r;
  const double r3 = r2 * r;
  const double r4 = r2 * r2;
  const double r5 = r3 * r2;
  const double r6 = r4 * r2;
  const double r7 = r5 * r2;
  const double sin_r = ((r + r3 * S3) + r5 * S5) + r7 * S7;
  const double cos_r = ((1.0 + r2 * C2) + r4 * C4) + r6 * C6;

  const unsigned m4 = (unsigned)(int)qm & 3u;           // |q| mod 4 (|q| small)
  const unsigned b0 = m4 & 1u;
  unsigned       b1 = (m4 >> 1) & 1u;
  b1 ^= b0 & (unsigned)(qs >> 63);                      // mod4_xor + mux on q sign

  sb = (uint64_t)__double_as_longlong(b0 ? cos_r : sin_r) ^ ((uint64_t)b1 << 63);
  cb = (uint64_t)__double_as_longlong(b0 ? sin_r : cos_r) ^ ((uint64_t)(b0 ^ b1) << 63);
}

// pulse k (k=0 MSB) <-> bit (63-k); chunk c holds pulses 4c..4c+3
__device__ __forceinline__ uint64_t decode_chunk(uint4 q, uint32_t c) {
  // pulses are exactly 0.0f/1.0f -> bit23 of the float pattern is the bit
  uint32_t nib = (((q.x >> 23) & 1u) << 3) | (((q.y >> 23) & 1u) << 2) |
                 (((q.z >> 23) & 1u) << 1) |  ((q.w >> 23) & 1u);
  return (uint64_t)nib << (60 - 4 * c);
}
__device__ __forceinline__ uint4 encode_chunk(uint64_t v, uint32_t c) {
  uint32_t nib = (uint32_t)(v >> (60 - 4 * c)) & 0xFu;
  uint4 w;
  w.x = ((nib >> 3) & 1u) * 0x3F800000u;
  w.y = ((nib >> 2) & 1u) * 0x3F800000u;
  w.z = ((nib >> 1) & 1u) * 0x3F800000u;
  w.w = ( nib       & 1u) * 0x3F800000u;
  return w;
}

// ---------------- tiled async-DMA kernel (full tiles only) ---------------
__global__ __launch_bounds__(THREADS)
void sincos_pulse_tiled(const float* __restrict__ in, float* __restrict__ out,
                        long long nElems) {
  __shared__ __align__(16) uint32_t lds[TILE_BYTES / 4];

  const uint32_t  tid      = threadIdx.x;
  const long long tileElem = (long long)blockIdx.x * TILE_ELEMS;
  const uint64_t  inT      = (uint64_t)(uintptr_t)(in + tileElem * 64);
  const uint64_t  sinT     = (uint64_t)(uintptr_t)(out + tileElem * 64);
  const uint64_t  cosT     = (uint64_t)(uintptr_t)(out + (nElems + tileElem) * 64);
  const uint32_t  ldsBase  = (uint32_t)(uintptr_t)(void*)lds;  // low 32 bits = LDS offset
  const uint32_t  rot      = tid & 15u;

  // ---- stage input tile: contiguous lane-major async copies (512B/wave/inst)
#pragma unroll
  for (int i = 0; i < 16; ++i) {
    const uint32_t L = tid * 16u + (uint32_t)i * 4096u;
    async_load16(ldsBase + L, inT + L);
  }
  wait_async0();
  __syncthreads();

  // ---- decode own element from LDS (chunk order rotated: bank-conflict-free)
  const uint4* lds4 = (const uint4*)lds;
  uint64_t xb = 0;
#pragma unroll
  for (int i = 0; i < 16; ++i) {
    const uint32_t c = ((uint32_t)i + rot) & 15u;
    xb |= decode_chunk(lds4[tid * 16u + c], c);
  }

  uint64_t sbits, cbits;
  sincos_fp64_bits(xb, sbits, cbits);

  __syncthreads();                 // everyone done reading input from LDS

  // ---- sin: element-major encode into LDS, then lane-major async store
  uint4* lds4w = (uint4*)lds;
#pragma unroll
  for (int i = 0; i < 16; ++i) {
    const uint32_t c = ((uint32_t)i + rot) & 15u;
    lds4w[tid * 16u + c] = encode_chunk(sbits, c);
  }
  __syncthreads();
#pragma unroll
  for (int i = 0; i < 16; ++i) {
    const uint32_t L = tid * 16u + (uint32_t)i * 4096u;
    async_store16(sinT + L, ldsBase + L);
  }
  wait_async0();                   // DMA done reading LDS (this wave)
  __syncthreads();                 // ...and every other wave too

  // ---- cos: reuse the same LDS buffer
#pragma unroll
  for (int i = 0; i < 16; ++i) {
    const uint32_t c = ((uint32_t)i + rot) & 15u;
    lds4w[tid * 16u + c] = encode_chunk(cbits, c);
  }
  __syncthreads();
#pragma unroll
  for (int i = 0; i < 16; ++i) {
    const uint32_t L = tid * 16u + (uint32_t)i * 4096u;
    async_store16(cosT + L, ldsBase + L);
  }
  wait_async0();
}

// ---------------- scalar fallback (any element count) --------------------
__global__ __launch_bounds__(THREADS)
void sincos_pulse_simple(const float* __restrict__ in, float* __restrict__ out,
                         long long nElems) {
  const long long e = (long long)blockIdx.x * blockDim.x + threadIdx.x;
  if (e >= nElems) return;
  const uint4* ip = (const uint4*)(in + e * 64);
  uint64_t xb = 0;
#pragma unroll
  for (uint32_t c = 0; c < 16; ++c) xb |= decode_chunk(ip[c], c);
  uint64_t sbits, cbits;
  sincos_fp64_bits(xb, sbits, cbits);
  uint4* sp = (uint4*)(out + e * 64);
  uint4* cp = (uint4*)(out + (nElems + e) * 64);
#pragma unroll
  for (uint32_t c = 0; c < 16; ++c) sp[c] = encode_chunk(sbits, c);
#pragma unroll
  for (uint32_t c = 0; c < 16; ++c) cp[c] = encode_chunk(cbits, c);
}

// ---------------- launcher ----------------------------------------------
extern "C" void kernel_launch(void* const* d_in, const int* in_sizes, int n_in,
                              void* d_out, int out_size, void* d_ws, size_t ws_size,
                              hipStream_t stream) {
  const float* in  = (const float*)d_in[0];
  float*       out = (float*)d_out;
  const long long totalPulses = (long long)in_sizes[0];
  const long long nElems      = totalPulses / 64;   // 512*2048 = 1,048,576

  if (nElems % TILE_ELEMS == 0) {
    const long long nTiles = nElems / TILE_ELEMS;   // 4096
    sincos_pulse_tiled<<<dim3((unsigned)nTiles), dim3(THREADS), 0, stream>>>(in, out, nElems);
  } else {
    const long long nBlocks = (nElems + THREADS - 1) / THREADS;
    sincos_pulse_simple<<<dim3((unsigned)nBlocks), dim3(THREADS), 0, stream>>>(in, out, nElems);
  }
}